// GCN_28424093565729
// MI455X (gfx1250) — compile-verified
//
#include <hip/hip_runtime.h>

typedef __attribute__((ext_vector_type(2))) float v2f;
typedef __attribute__((ext_vector_type(8))) float v8f;

#define N_NODES 100000
#define N_EDGES 3200000
#define D_FEAT  256
#define HIDDEN  64
#define NCLASS  7
#define ROWBLKS (N_NODES / 16)   // 6250, exact

// ---------------- degree / normalization ----------------

__global__ void k_zero(float* __restrict__ p, int n) {
    int i = blockIdx.x * blockDim.x + threadIdx.x;
    if (i < n) p[i] = 0.0f;
}

__global__ void k_deg(const int* __restrict__ ei, const float* __restrict__ ew,
                      float* __restrict__ deg) {
    int e = blockIdx.x * blockDim.x + threadIdx.x;
    if (e < N_EDGES) atomicAdd(&deg[ei[N_EDGES + e]], ew[e]);
}

__global__ void k_dinv(float* __restrict__ deg) {
    int i = blockIdx.x * blockDim.x + threadIdx.x;
    if (i < N_NODES) {
        float d = deg[i] + 1.0f;             // self-loop weight 1
        deg[i] = rsqrtf(fmaxf(d, 1e-12f));   // in place: deg -> dinv
    }
}

// ---------------- GEMM0: m0 = x @ W0  (fp32 WMMA 16x16x4) ----------------
// block = 128 threads = 4 waves; block handles 16 rows x 64 cols;
// wave w computes 16x16 tile at columns [16w,16w+16).

__global__ __launch_bounds__(128) void k_gemm0(const float* __restrict__ x,
                                               const float* __restrict__ W0,
                                               float* __restrict__ m0) {
    const int lane = threadIdx.x & 31;
    const int wv   = threadIdx.x >> 5;
    const int row0 = blockIdx.x * 16;
    const int col0 = wv * 16;
    const int hh   = lane >> 4;     // 0: K pair {0,1}; 1: K pair {2,3}
    const int l15  = lane & 15;

    v8f c = {};
    const float* xrow = x  + (row0 + l15) * D_FEAT + 2 * hh;
    const float* wcol = W0 + (2 * hh) * HIDDEN + col0 + l15;

#pragma unroll 4
    for (int k = 0; k < D_FEAT; k += 4) {
        v2f a, b;
        a.x = xrow[k];                    // (M=l15, K=k+2hh)
        a.y = xrow[k + 1];                // (M=l15, K=k+2hh+1)
        b.x = wcol[k * HIDDEN];           // (K=k+2hh,   N=col0+l15)
        b.y = wcol[k * HIDDEN + HIDDEN];  // (K=k+2hh+1, N=col0+l15)
        c = __builtin_amdgcn_wmma_f32_16x16x4_f32(false, a, false, b,
                                                  (short)0, c, false, false);
    }
    // D layout: VGPR v -> M = v + 8*hh, N = l15
    float* out = m0 + (row0 + 8 * hh) * HIDDEN + col0 + l15;
#pragma unroll
    for (int v = 0; v < 8; ++v) out[v * HIDDEN] = c[v];
}

// ---------------- self-loop init + edge scatter (layer 0) ----------------

__global__ void k_init_h(const float* __restrict__ dinv, const float* __restrict__ m0,
                         float* __restrict__ h) {
    int idx = blockIdx.x * blockDim.x + threadIdx.x;
    if (idx < N_NODES * HIDDEN) {
        float di = dinv[idx / HIDDEN];
        h[idx] = di * di * m0[idx];       // self-loop contribution, w=1
    }
}

// one wave32 per edge; each lane handles 2 consecutive features (float2 gather)
__global__ __launch_bounds__(256) void k_scatter0(const int* __restrict__ ei,
                                                  const float* __restrict__ ew,
                                                  const float* __restrict__ dinv,
                                                  const float* __restrict__ m0,
                                                  float* __restrict__ h) {
    const int lane = threadIdx.x & 31;
    const int e = blockIdx.x * 8 + (threadIdx.x >> 5);
    if (e >= N_EDGES) return;
    const int s = ei[e];
    const int d = ei[N_EDGES + e];
    const float wn = dinv[s] * ew[e] * dinv[d];
    const float2 v = *(const float2*)(m0 + s * HIDDEN + 2 * lane);
    atomicAdd(&h[d * HIDDEN + 2 * lane],     wn * v.x);
    atomicAdd(&h[d * HIDDEN + 2 * lane + 1], wn * v.y);
}

__global__ void k_bias_relu(float* __restrict__ h, const float* __restrict__ b0) {
    int idx = blockIdx.x * blockDim.x + threadIdx.x;
    if (idx < N_NODES * HIDDEN)
        h[idx] = fmaxf(h[idx] + b0[idx % HIDDEN], 0.0f);
}

// ---------------- GEMM1: m1 = h @ W1 (N=7 padded to 16-wide tile) ----------------

__global__ __launch_bounds__(128) void k_gemm1(const float* __restrict__ h,
                                               const float* __restrict__ W1,
                                               float* __restrict__ m1) {
    const int lane = threadIdx.x & 31;
    const int wv   = threadIdx.x >> 5;
    const int rb   = blockIdx.x * 4 + wv;
    if (rb >= ROWBLKS) return;            // whole-wave exit only
    const int row0 = rb * 16;
    const int hh   = lane >> 4;
    const int l15  = lane & 15;

    v8f c = {};
    const float* hrow = h + (row0 + l15) * HIDDEN + 2 * hh;
#pragma unroll
    for (int k = 0; k < HIDDEN; k += 4) {
        v2f a, b;
        a.x = hrow[k];
        a.y = hrow[k + 1];
        float bx = 0.0f, by = 0.0f;
        if (l15 < NCLASS) {               // zero-pad columns 7..15
            bx = W1[(k + 2 * hh) * NCLASS + l15];
            by = W1[(k + 2 * hh + 1) * NCLASS + l15];
        }
        b.x = bx; b.y = by;
        c = __builtin_amdgcn_wmma_f32_16x16x4_f32(false, a, false, b,
                                                  (short)0, c, false, false);
    }
    if (l15 < NCLASS) {
        float* out = m1 + (row0 + 8 * hh) * NCLASS + l15;
#pragma unroll
        for (int v = 0; v < 8; ++v) out[v * NCLASS] = c[v];
    }
}

// ---------------- layer-1 self-loop/bias init + scatter ----------------

__global__ void k_init_out(const float* __restrict__ dinv, const float* __restrict__ m1,
                           const float* __restrict__ b1, float* __restrict__ out) {
    int idx = blockIdx.x * blockDim.x + threadIdx.x;
    if (idx < N_NODES * NCLASS) {
        int i = idx / NCLASS, j = idx - i * NCLASS;
        float di = dinv[i];
        out[idx] = di * di * m1[idx] + b1[j];
    }
}

// 8 lanes per edge (lane j<7 active); 32 edges per 256-thread block
__global__ __launch_bounds__(256) void k_scatter1(const int* __restrict__ ei,
                                                  const float* __restrict__ ew,
                                                  const float* __restrict__ dinv,
                                                  const float* __restrict__ m1,
                                                  float* __restrict__ out) {
    const int j = threadIdx.x & 7;
    const int e = blockIdx.x * 32 + (threadIdx.x >> 3);
    if (e >= N_EDGES || j >= NCLASS) return;
    const int s = ei[e];
    const int d = ei[N_EDGES + e];
    const float wn = dinv[s] * ew[e] * dinv[d];
    atomicAdd(&out[d * NCLASS + j], wn * m1[s * NCLASS + j]);
}

// ---------------- launch ----------------

extern "C" void kernel_launch(void* const* d_in, const int* in_sizes, int n_in,
                              void* d_out, int out_size, void* d_ws, size_t ws_size,
                              hipStream_t stream) {
    const float* x  = (const float*)d_in[0];
    const int*   ei = (const int*)  d_in[1];   // [2, E]
    const float* ew = (const float*)d_in[2];
    const float* W0 = (const float*)d_in[3];
    const float* b0 = (const float*)d_in[4];
    const float* W1 = (const float*)d_in[5];
    const float* b1 = (const float*)d_in[6];
    float* out = (float*)d_out;

    float* ws   = (float*)d_ws;
    float* dinv = ws;                              // N
    float* m0   = dinv + N_NODES;                  // N*64
    float* h    = m0 + (size_t)N_NODES * HIDDEN;   // N*64
    float* m1   = h  + (size_t)N_NODES * HIDDEN;   // N*7

    // 1) degree -> dinv
    k_zero<<<(N_NODES + 255) / 256, 256, 0, stream>>>(dinv, N_NODES);
    k_deg <<<(N_EDGES + 255) / 256, 256, 0, stream>>>(ei, ew, dinv);
    k_dinv<<<(N_NODES + 255) / 256, 256, 0, stream>>>(dinv);

    // 2) m0 = x @ W0
    k_gemm0<<<ROWBLKS, 128, 0, stream>>>(x, W0, m0);

    // 3) layer-0 aggregation: self-loops init, then edge scatter, bias+relu
    k_init_h<<<(N_NODES * HIDDEN + 255) / 256, 256, 0, stream>>>(dinv, m0, h);
    k_scatter0<<<(N_EDGES + 7) / 8, 256, 0, stream>>>(ei, ew, dinv, m0, h);
    k_bias_relu<<<(N_NODES * HIDDEN + 255) / 256, 256, 0, stream>>>(h, b0);

    // 4) m1 = h @ W1
    k_gemm1<<<(ROWBLKS + 3) / 4, 128, 0, stream>>>(h, W1, m1);

    // 5) layer-1 aggregation into d_out
    k_init_out<<<(N_NODES * NCLASS + 255) / 256, 256, 0, stream>>>(dinv, m1, b1, out);
    k_scatter1<<<(N_EDGES + 31) / 32, 256, 0, stream>>>(ei, ew, dinv, m1, out);
}